// XAJMul_6270652252535
// MI455X (gfx1250) — compile-verified
//
#include <hip/hip_runtime.h>
#include <stdint.h>

// XAJ hydrological scan for MI455X (gfx1250).
// - Pure elementwise recurrence: no matmul FLOPs -> WMMA inapplicable.
// - CDNA5 path used: async global->LDS staging (ASYNCcnt) with double
//   buffering via inline "global_load_async_to_lds_b128" + "s_wait_asynccnt",
//   plus wave-level 8-lane reductions (ds path) for the mu-ensemble means.

#define PRECS 1e-6f
#define GPB   8                 // grid cells per block
#define CH    32                // timesteps staged per chunk
#define ROWF  (GPB * 3)         // floats per staged t-row (P, mid, PET per grid) = 24
#define BUFF  (CH * ROWF)       // floats per LDS buffer = 768 (3 KB)

__device__ __forceinline__ float fpowf_fast(float x, float y) {
    // x >= 0 always here; log2(0) = -inf -> exp2(-inf) = 0, matching pow(0, +e) = 0
    // v_log_f32 is log2; v_exp_f32 is exp2.
    return __builtin_amdgcn_exp2f(y * __builtin_amdgcn_logf(x));
}
__device__ __forceinline__ float frcp(float x) { return __builtin_amdgcn_rcpf(x); }

__global__ __launch_bounds__(64) void xaj_scan_kernel(
    const float* __restrict__ x,       // (Nstep, Ngrid, 3)  [P, _, PET]
    const float* __restrict__ params,  // (Ngrid, 12, 8)
    float* __restrict__ out,           // (Nstep, Ngrid, 5)
    int Nstep, int Ngrid)
{
    __shared__ float sx[2 * BUFF];

    const int tid = threadIdx.x;       // 0..63
    const int g0  = blockIdx.x * GPB;  // first grid of this block
    const int gl  = tid >> 3;          // local grid 0..7
    const int mu  = tid & 7;           // ensemble member 0..7
    const int g   = g0 + gl;
    const bool active = (g < Ngrid);
    const int gc = active ? g : (Ngrid - 1);

    // ---------------- parameters: load once, scale, hoist invariants ----------
    const float lo[12] = {0.3f, 0.0f, 0.01f, 0.01f, 0.01f, 0.09f,
                          0.01f, 0.0f, 0.01f, 0.01f, 0.0f, 0.0f};
    const float hi[12] = {2.0f, 5.0f, 100.f, 100.f, 200.f, 1.0f,
                          100.f, 10.f, 0.7f, 0.7f, 0.998f, 0.998f};
    const float* pp = params + (size_t)gc * 96 + mu;
    float pv[12];
#pragma unroll
    for (int i = 0; i < 12; ++i)
        pv[i] = lo[i] + pp[i * 8] * (hi[i] - lo[i]);

    const float KE = pv[0], B = pv[1], WUM = pv[2], WLM = pv[3], WM = pv[4],
                C = pv[5], SM = pv[6], EX = pv[7], KIr = pv[8], KGr = pv[9],
                CI = pv[10], CG = pv[11];

    const float wdm  = fmaxf(WM - WUM - WLM, 0.f);
    const float sumk = KIr + KGr;
    const float adj  = (1.f - PRECS) * frcp(sumk + PRECS);
    const float KI   = (sumk < 1.f) ? KIr : KIr * adj;
    const float KG   = (sumk < 1.f) ? KGr : KGr * adj;
    const float WMM = WM * (1.f + B);
    const float SMM = SM * (1.f + EX);
    const float invWLM = frcp(WLM + PRECS);
    const float invWM  = frcp(WM + PRECS);
    const float invWMM = frcp(WMM + PRECS);
    const float invSM  = frcp(SM + PRECS);
    const float invSMM = frcp(SMM + PRECS);
    const float expB  = frcp(1.f + B);
    const float oneB  = 1.f + B;
    const float expEX = frcp(1.f + EX);
    const float oneEX = 1.f + EX;
    const float keepS = 1.f - KI - KG;
    const float omCI = 1.f - CI, omCG = 1.f - CG;

    // state
    float WU = 1e-3f, WL = 1e-3f, WD = 1e-3f, S = 1e-3f,
          FR0 = 1e-3f, QI = 1e-3f, QG = 1e-3f;

    // ---------------- async staging setup -------------------------------------
    // LDS byte offset of sx: low 32 bits of the flat shared-aperture address.
    const uint32_t ldsBase = (uint32_t)(uintptr_t)(&sx[0]);
    const uint64_t rowBytes = (uint64_t)Ngrid * 3u * 4u;          // one t-row of x
    const uint64_t xBase  = (uint64_t)(uintptr_t)x;
    const uint64_t xClamp = xBase + (uint64_t)Nstep * rowBytes - 16u; // last safe b128
    const uint64_t xBlock = xBase + (uint64_t)g0 * 12u;           // block's row offset
    const int nch = (Nstep + CH - 1) / CH;

    // Stage chunk c into buffer buf: 32 rows x 96B = 192 b128 transfers,
    // exactly 3 per lane (addresses clamped -> ASYNCcnt bookkeeping constant).
    auto stage = [&](int c, int buf) {
        const int t0 = c * CH;
        const uint32_t bb = ldsBase + (uint32_t)buf * (BUFF * 4);
#pragma unroll
        for (int k = 0; k < 3; ++k) {
            int i = tid + 64 * k;            // transfer index 0..191
            int t = i / 6, seg = i - t * 6;  // row 0..31, 16B segment 0..5
            int tg = t0 + t; if (tg > Nstep - 1) tg = Nstep - 1;
            uint64_t ga = xBlock + (uint64_t)tg * rowBytes + (uint32_t)(seg * 16);
            if (ga > xClamp) ga = xClamp;    // keep strictly in-bounds
            uint32_t la = bb + (uint32_t)(t * (ROWF * 4) + seg * 16);
            asm volatile("global_load_async_to_lds_b128 %0, %1, off"
                         :: "v"(la), "v"(ga) : "memory");
        }
    };

    stage(0, 0);

    // ---------------- main scan ------------------------------------------------
    for (int c = 0; c < nch; ++c) {
        if (c + 1 < nch) {
            stage(c + 1, (c + 1) & 1);
            // 3 newest loads may remain in flight; the 3 for chunk c are done.
            asm volatile("s_wait_asynccnt 0x3" ::: "memory");
        } else {
            asm volatile("s_wait_asynccnt 0x0" ::: "memory");
        }
        __syncthreads();

        const float* buf = &sx[(c & 1) * BUFF];
        const int t0 = c * CH;
        int steps = Nstep - t0; if (steps > CH) steps = CH;

        for (int t = 0; t < steps; ++t) {
            const float P_t  = buf[t * ROWF + gl * 3 + 0];
            const float PETr = buf[t * ROWF + gl * 3 + 2];

            const float PET_t = PETr * KE;
            const float EU = fminf(PET_t, WU + P_t);
            const float D  = fmaxf(PET_t - EU, 0.f);
            const float wet_excess = fmaxf(WL - C * WLM, 0.f);
            const float wet_ratio  = wet_excess * frcp(wet_excess + 1.f);
            const float EL_wet = D * WL * invWLM;
            const float EL_dry = fminf(C * D, WL);
            float EL = fmaxf(wet_ratio * EL_wet + (1.f - wet_ratio) * EL_dry, 0.f);
            EL = fminf(EL, WL);
            const float D_res = fmaxf(D - EL, 0.f);
            const float ED = fminf(C * D_res, WD);
            const float E  = EU + EL + ED;
            const float PE = fmaxf(P_t - E, 0.f);
            const float W  = WU + WL + WD;
            const float base_W = fmaxf(1.f - W * invWM, PRECS);
            const float A = WMM * (1.f - fpowf_fast(base_W, expB));
            const float ratio = fminf((PE + A) * invWMM, 1.f);
            float R_per = (PE + A < WMM)
                ? (PE - (WM - W) + WM * fpowf_fast(1.f - ratio, oneB))
                : (PE - (WM - W));
            R_per = fmaxf(R_per, 0.f);

            const float dW = P_t - E - R_per;
            float dpos = fmaxf(dW, 0.f);
            float f1 = fminf(dpos, fmaxf(WUM - WU, 0.f)); WU += f1; dpos -= f1;
            float f2 = fminf(dpos, fmaxf(WLM - WL, 0.f)); WL += f2; dpos -= f2;
            float f3 = fminf(dpos, fmaxf(wdm - WD, 0.f)); WD += f3;
            float dneg = fmaxf(-dW, 0.f);
            float s1 = fminf(dneg, WU); WU -= s1; dneg -= s1;
            float s2 = fminf(dneg, WL); WL -= s2; dneg -= s2;
            float s3 = fminf(dneg, WD); WD -= s3;

            float FR = (PE > 0.f) ? (R_per * frcp(PE + PRECS)) : FR0;
            FR = fminf(fmaxf(FR, 0.f), 1.f);
            const float invFR = frcp(FR + PRECS);
            const float S_eq = fminf(S * FR0 * invFR, SM);
            const float ratio_s = fmaxf(1.f - S_eq * invSM, PRECS);
            const float AU = SMM * (1.f - fpowf_fast(ratio_s, expEX));
            const float ratio_pa = fminf((PE + AU) * invSMM, 1.f);
            float RS = (PE + AU < SMM)
                ? ((PE + S_eq - SM + SM * fpowf_fast(1.f - ratio_pa, oneEX)) * FR)
                : ((PE + S_eq - SM) * FR);
            RS = fminf(fmaxf(RS, 0.f), R_per);
            S = fminf(S_eq + (R_per - RS) * invFR, SM);
            const float RI = KI * S * FR, RG = KG * S * FR;
            S *= keepS;
            QI = CI * QI + omCI * RI;
            QG = CG * QG + omCG * RG;
            FR0 = FR;

            // --- mu-ensemble means over 8 adjacent lanes (wave32, width-8 xor)
            float sRS = RS, sQI = QI, sQG = QG, sE = E;
#pragma unroll
            for (int m = 1; m < 8; m <<= 1) {
                sRS += __shfl_xor(sRS, m, 8);
                sQI += __shfl_xor(sQI, m, 8);
                sQG += __shfl_xor(sQG, m, 8);
                sE  += __shfl_xor(sE,  m, 8);
            }
            if (mu == 0 && active) {
                float* o = out + ((size_t)(t0 + t) * Ngrid + g) * 5;
                const float inv8 = 0.125f;
                o[0] = (sRS + sQI + sQG) * inv8;  // Qs
                o[1] = sRS * inv8;                // QSave
                o[2] = sQI * inv8;                // QIave
                o[3] = sQG * inv8;                // QGave
                o[4] = sE  * inv8;                // ETave
            }
        }
        __syncthreads();  // protect buffer reuse before next stage()
    }
}

extern "C" void kernel_launch(void* const* d_in, const int* in_sizes, int n_in,
                              void* d_out, int out_size, void* d_ws, size_t ws_size,
                              hipStream_t stream) {
    const float* x      = (const float*)d_in[0];   // (Nstep, Ngrid, 3) f32
    const float* params = (const float*)d_in[1];   // (Ngrid, 12, 8)   f32
    const int mu = 8;                              // reference setup
    const int Ngrid = in_sizes[1] / (12 * mu);     // 2000
    const int Nstep = in_sizes[0] / (Ngrid * 3);   // 365
    const int nblocks = (Ngrid + GPB - 1) / GPB;   // 250
    xaj_scan_kernel<<<nblocks, 64, 0, stream>>>(x, params, (float*)d_out,
                                                Nstep, Ngrid);
}